// BertLinearCRF_42365557408234
// MI455X (gfx1250) — compile-verified
//
#include <hip/hip_runtime.h>
#include <hip/hip_bf16.h>
#include <math.h>

typedef __attribute__((ext_vector_type(2))) float v2f;
typedef __attribute__((ext_vector_type(4))) float v4f;
typedef __attribute__((ext_vector_type(8))) float v8f;

#define BB   32
#define SS   512
#define HH   1024
#define LL   17
#define MTOT (BB * SS)          // 16384 rows
#define KCHUNK 256              // K per LDS chunk
#define KSTEPS (KCHUNK / 4)     // 64 wmma k-steps per chunk
#define GRP   8                 // k-steps per load group
#define NGRP  (KSTEPS / GRP)    // 8 groups per chunk

// ---------------------------------------------------------------------------
// Kernel 1: emissions = hidden_states @ W^T + b   via V_WMMA_F32_16X16X4_F32
// Grid: 128 blocks x 256 threads (8 waves); each wave owns a 16-row M tile,
// computes two 16-col N tiles (cols 0..15, 16..31; only 0..16 stored).
// B (=W^T) fragments are pre-packed into LDS in the exact wave32 WMMA layout.
// Inner loop: groups of 8 k-steps with ping-pong register arrays so all 8
// global/LDS loads of the next group are in flight (distinct live registers,
// the allocator cannot coalesce them) while the current group's WMMAs run.
// ---------------------------------------------------------------------------
__global__ __launch_bounds__(256)
void emissions_wmma_kernel(const float* __restrict__ hs,
                           const float* __restrict__ W,
                           const float* __restrict__ bias,
                           float* __restrict__ em)
{
    __shared__ __align__(16) float sB[KSTEPS * 128];   // 32 KB

    const int tid    = threadIdx.x;
    const int wave   = tid >> 5;
    const int lane   = tid & 31;
    const int lane16 = lane & 15;
    const int laneHi = lane >> 4;               // 0 or 1
    const int tile   = blockIdx.x * 8 + wave;   // 0..1023
    const int rowBase = tile * 16;

    // init accumulators with bias
    v8f c0, c1;
    {
        const float b0 = bias[lane16];
        const float b1 = (lane16 == 0) ? bias[16] : 0.0f;
        #pragma unroll
        for (int v = 0; v < 8; ++v) { c0[v] = b0; c1[v] = b1; }
    }

    const float* aRow = hs + (size_t)(rowBase + lane16) * HH + (laneHi << 1);

    for (int kc = 0; kc < HH / KCHUNK; ++kc) {
        const int kBase = kc * KCHUNK;

        __syncthreads();
        // cooperative fill of B fragments (zero-padded cols 17..31)
        #pragma unroll
        for (int i = 0; i < (KSTEPS * 128) / 256; ++i) {
            const int e     = tid + i * 256;
            const int kstep = e >> 7;
            const int r     = e & 127;
            const int fl    = r >> 2;           // fragment lane
            const int j     = r & 3;            // 0,1: tile0 v0,v1  2,3: tile1 v0,v1
            const int n     = (fl & 15) + ((j >> 1) << 4);
            const int k     = kBase + (kstep << 2) + ((fl >> 4) << 1) + (j & 1);
            sB[e] = (n < LL) ? W[n * HH + k] : 0.0f;
        }
        __syncthreads();

        const float* aPtr = aRow + kBase;

        // ping-pong buffers: all elements simultaneously live -> distinct VGPRs
        v2f aB[2][GRP];
        v4f fB[2][GRP];

        #pragma unroll
        for (int s = 0; s < GRP; ++s) {
            aB[0][s] = *reinterpret_cast<const v2f*>(aPtr + (s << 2));
            fB[0][s] = *reinterpret_cast<const v4f*>(&sB[(s << 7) + (lane << 2)]);
        }

        #pragma unroll
        for (int g = 0; g < NGRP; ++g) {
            const int cur = g & 1;
            const int nxt = cur ^ 1;
            if (g + 1 < NGRP) {
                #pragma unroll
                for (int s = 0; s < GRP; ++s) {
                    const int ks = (g + 1) * GRP + s;
                    aB[nxt][s] = *reinterpret_cast<const v2f*>(aPtr + (ks << 2));
                    fB[nxt][s] = *reinterpret_cast<const v4f*>(
                                     &sB[(ks << 7) + (lane << 2)]);
                }
            }
            #pragma unroll
            for (int s = 0; s < GRP; ++s) {
                v2f b0; b0.x = fB[cur][s].x; b0.y = fB[cur][s].y;
                v2f b1; b1.x = fB[cur][s].z; b1.y = fB[cur][s].w;
                c0 = __builtin_amdgcn_wmma_f32_16x16x4_f32(false, aB[cur][s], false, b0,
                                                           (short)0, c0, false, false);
                c1 = __builtin_amdgcn_wmma_f32_16x16x4_f32(false, aB[cur][s], false, b1,
                                                           (short)0, c1, false, false);
            }
        }
    }

    // store C/D: vgpr v, lanes0-15 -> M=v, lanes16-31 -> M=v+8; N = lane%16
    #pragma unroll
    for (int v = 0; v < 8; ++v) {
        const int m = rowBase + v + (laneHi << 3);
        em[(size_t)m * LL + lane16] = c0[v];
        if (lane16 == 0) em[(size_t)m * LL + 16] = c1[v];
    }
}

// ---------------------------------------------------------------------------
// Kernel 2: CRF numerator + forward scan. One wave per batch element.
// ---------------------------------------------------------------------------
__global__ __launch_bounds__(32)
void crf_kernel(const float* __restrict__ em,
                const int*   __restrict__ labels,
                const float* __restrict__ startT,
                const float* __restrict__ endT,
                const float* __restrict__ trans,
                float* __restrict__ llh)
{
    __shared__ float sTrans[LL * LL];
    __shared__ float sAlpha[32];
    __shared__ float sStart[LL];
    __shared__ float sEnd[LL];

    const int b   = blockIdx.x;
    const int tid = threadIdx.x;

    for (int i = tid; i < LL * LL; i += 32) sTrans[i] = trans[i];
    if (tid < LL) { sStart[tid] = startT[tid]; sEnd[tid] = endT[tid]; }
    __syncthreads();

    const int*   lab = labels + (size_t)b * SS;
    const float* emb = em + (size_t)b * SS * LL;

    // ---- numerator (parallel over t, wave reduction) ----
    float part = 0.0f;
    int   cnt  = 0;
    for (int t = tid; t < SS; t += 32) {
        const int l  = lab[t];
        const bool m = (l != -100);
        const int tg = m ? l : 0;
        cnt += m ? 1 : 0;
        if (t == 0) {
            part += sStart[tg] + emb[tg];
        } else if (m) {
            const int lp  = lab[t - 1];
            const int tgp = (lp != -100) ? lp : 0;
            part += sTrans[tgp * LL + tg] + emb[t * LL + tg];
        }
    }
    #pragma unroll
    for (int off = 16; off > 0; off >>= 1) {
        part += __shfl_xor(part, off, 32);
        cnt  += __shfl_xor(cnt, off, 32);
    }
    float numerator = 0.0f;
    if (tid == 0) {
        const int se = cnt - 1;
        const int ll = lab[se];
        const int lt = (ll != -100) ? ll : 0;
        numerator = part + sEnd[lt];
    }

    // ---- forward scan ----
    float colT[LL];   // transitions column for my next-state j = tid
    if (tid < LL) {
        #pragma unroll
        for (int i = 0; i < LL; ++i) colT[i] = sTrans[i * LL + tid];
        sAlpha[tid] = sStart[tid] + emb[tid];
    }
    __syncthreads();

    for (int t = 1; t < SS; ++t) {
        const int  l = lab[t];          // uniform -> scalar load
        const bool m = (l != -100);
        float nxt = 0.0f;
        float av[LL];
        if (tid < LL) {
            float mx = -3.4e38f;
            #pragma unroll
            for (int i = 0; i < LL; ++i) {
                const float v = sAlpha[i] + colT[i];
                av[i] = v;
                mx = fmaxf(mx, v);
            }
            float s = 0.0f;
            #pragma unroll
            for (int i = 0; i < LL; ++i) s += __expf(av[i] - mx);
            nxt = emb[t * LL + tid] + mx + __logf(s);
        }
        __syncthreads();
        if (tid < LL && m) sAlpha[tid] = nxt;
        __syncthreads();
    }

    if (tid == 0) {
        float mx = -3.4e38f;
        #pragma unroll
        for (int j = 0; j < LL; ++j) mx = fmaxf(mx, sAlpha[j] + sEnd[j]);
        float s = 0.0f;
        #pragma unroll
        for (int j = 0; j < LL; ++j) s += __expf(sAlpha[j] + sEnd[j] - mx);
        const float partition = mx + __logf(s);
        llh[b] = numerator - partition;
    }
}

// ---------------------------------------------------------------------------
// Kernel 3: loss = -mean(llh)
// ---------------------------------------------------------------------------
__global__ __launch_bounds__(32)
void loss_kernel(const float* __restrict__ llh, float* __restrict__ out)
{
    const int tid = threadIdx.x;
    float v = llh[tid];
    #pragma unroll
    for (int off = 16; off > 0; off >>= 1) v += __shfl_xor(v, off, 32);
    if (tid == 0) out[0] = -(v / (float)BB);
}

// ---------------------------------------------------------------------------
extern "C" void kernel_launch(void* const* d_in, const int* in_sizes, int n_in,
                              void* d_out, int out_size, void* d_ws, size_t ws_size,
                              hipStream_t stream)
{
    const float* hs     = (const float*)d_in[0];   // (B,S,H)
    // d_in[1] attention_mask: unused by the reference math
    const int*   labels = (const int*)d_in[2];     // (B,S)
    const float* W      = (const float*)d_in[3];   // (L,H)
    const float* bias   = (const float*)d_in[4];   // (L,)
    const float* startT = (const float*)d_in[5];   // (L,)
    const float* endT   = (const float*)d_in[6];   // (L,)
    const float* trans  = (const float*)d_in[7];   // (L,L)

    float* out = (float*)d_out;          // [0] = loss, [1..] = emissions (B,S,L)
    float* em  = out + 1;
    float* llh = (float*)d_ws;           // 32 floats of scratch

    emissions_wmma_kernel<<<MTOT / (16 * 8), 256, 0, stream>>>(hs, W, bias, em);
    crf_kernel<<<BB, 32, 0, stream>>>(em, labels, startT, endT, trans, llh);
    loss_kernel<<<1, 32, 0, stream>>>(llh, out);
}